// GradTTS_69217692942363
// MI455X (gfx1250) — compile-verified
//
#include <hip/hip_runtime.h>
#include <math.h>

#define B_    32
#define TX    512
#define TY    1024
#define DD    256
#define NMEL  80
#define NSPK  64
#define NEGF  (-1e9f)
#define LOG2PI 1.8378770664093453f

typedef __attribute__((ext_vector_type(2))) float v2f;
typedef __attribute__((ext_vector_type(8))) float v8f;

__device__ __forceinline__ v8f wmma4(v2f a, v2f b, v8f c) {
  // V_WMMA_F32_16X16X4_F32 : D(16x16 f32) = A(16x4) * B(4x16) + C
  return __builtin_amdgcn_wmma_f32_16x16x4_f32(
      /*neg_a=*/false, a, /*neg_b=*/false, b,
      /*c_mod=*/(short)0, c, /*reuse_a=*/false, /*reuse_b=*/false);
}

// ---------------------------------------------------------------------------
// Kernel A: features[m,n] = mask * (contents[m,:] @ W_text[:,n]) + spk_emb
// M = B*TX = 16384, N = K = 256. One 16x16 C tile per wave, 8 waves/block.
// ---------------------------------------------------------------------------
__global__ void k_features(const float* __restrict__ A, const float* __restrict__ Bw,
                           const int* __restrict__ spk, const float* __restrict__ emb,
                           const int* __restrict__ xlens, float* __restrict__ feat) {
  const int lane = threadIdx.x & 31;
  const int wave = threadIdx.x >> 5;
  const int tile = blockIdx.x * 8 + wave;          // 16384 tiles
  const int mt = tile >> 4, nt = tile & 15;
  const int m0 = mt * 16, n0 = nt * 16;
  const int row = lane & 15, kh = lane >> 4;       // kh selects K half (ISA A/B layout)
  const float* Ap = A + (size_t)(m0 + row) * DD + kh * 2;
  const float* Bp = Bw + n0 + row + (size_t)(kh * 2) * DD;
  v8f acc = {};
  for (int k = 0; k < DD; k += 4) {
    v2f a, b;
    a.x = Ap[k];               a.y = Ap[k + 1];
    b.x = Bp[(size_t)k * DD];  b.y = Bp[(size_t)k * DD + DD];
    acc = wmma4(a, b, acc);
  }
  const int bb = m0 >> 9;                          // TX = 512 divides tile rows
  const int xl = xlens[bb];
  const int s  = spk[bb];
  const int n  = n0 + row;
  const float e = emb[(size_t)s * DD + n];
  for (int r = 0; r < 8; ++r) {
    int m = m0 + r + kh * 8;                       // C layout: M = r + 8*kh
    int x = m & (TX - 1);
    float v = acc[r];
    if (x >= xl) v = 0.f;                          // mask BEFORE adding spk_emb (ref order)
    feat[(size_t)m * DD + n] = v + e;
  }
}

// ---------------------------------------------------------------------------
// Kernel B: mu_x[m,c] = mask * (features[m,:] @ W_mel[:,c]); stored (B,Tx,80)
// ---------------------------------------------------------------------------
__global__ void k_mux(const float* __restrict__ F, const float* __restrict__ Bw,
                      const int* __restrict__ xlens, float* __restrict__ mux) {
  const int lane = threadIdx.x & 31;
  const int wave = threadIdx.x >> 5;
  const int tile = blockIdx.x * 8 + wave;          // 1024*5 = 5120 tiles
  const int mt = tile / 5, nt = tile % 5;
  const int m0 = mt * 16, n0 = nt * 16;
  const int row = lane & 15, kh = lane >> 4;
  const float* Ap = F + (size_t)(m0 + row) * DD + kh * 2;
  const float* Bp = Bw + n0 + row + (size_t)(kh * 2) * NMEL;
  v8f acc = {};
  for (int k = 0; k < DD; k += 4) {
    v2f a, b;
    a.x = Ap[k];                 a.y = Ap[k + 1];
    b.x = Bp[(size_t)k * NMEL];  b.y = Bp[(size_t)k * NMEL + NMEL];
    acc = wmma4(a, b, acc);
  }
  const int bb = m0 >> 9;
  const int xl = xlens[bb];
  const int n  = n0 + row;
  for (int r = 0; r < 8; ++r) {
    int m = m0 + r + kh * 8;
    int x = m & (TX - 1);
    float v = acc[r];
    if (x >= xl) v = 0.f;
    mux[(size_t)m * NMEL + n] = v;
  }
}

// ---------------------------------------------------------------------------
// Row ops: logw[m] = mask * <features[m,:], W_dur>, musq[m] = -0.5*||mu_x[m]||^2
// ---------------------------------------------------------------------------
__global__ void k_rowops(const float* __restrict__ F, const float* __restrict__ mux,
                         const float* __restrict__ Wd, const int* __restrict__ xlens,
                         float* __restrict__ logw, float* __restrict__ musq) {
  int m = blockIdx.x * blockDim.x + threadIdx.x;
  if (m >= B_ * TX) return;
  int bb = m >> 9, x = m & (TX - 1);
  const float* fr = F + (size_t)m * DD;
  float s = 0.f;
  for (int k = 0; k < DD; ++k) s += fr[k] * Wd[k];
  logw[m] = (x < xlens[bb]) ? s : 0.f;
  const float* mr = mux + (size_t)m * NMEL;
  float q = 0.f;
  for (int c = 0; c < NMEL; ++c) q += mr[c] * mr[c];
  musq[m] = -0.5f * q;
}

__global__ void k_ysq(const float* __restrict__ mel, float* __restrict__ ysq) {
  int m = blockIdx.x * blockDim.x + threadIdx.x;
  if (m >= B_ * TY) return;
  const float* mr = mel + (size_t)m * NMEL;
  float q = 0.f;
  for (int c = 0; c < NMEL; ++c) q += mr[c] * mr[c];
  ysq[m] = -0.5f * q;
}

// ---------------------------------------------------------------------------
// Kernel C: log_prior transposed, LP[b,y,x] = ysq + <mel[b,y],mu_x[b,x]> + musq
//           + const ; NEG where masked. Batched WMMA GEMM, K = 80.
// ---------------------------------------------------------------------------
__global__ void k_logprior(const float* __restrict__ mel, const float* __restrict__ mux,
                           const float* __restrict__ ysq, const float* __restrict__ musq,
                           const int* __restrict__ xlens, const int* __restrict__ ylens,
                           float* __restrict__ LP) {
  const int lane = threadIdx.x & 31;
  const int wave = threadIdx.x >> 5;
  const int tile = blockIdx.x * 8 + wave;          // 32 * 64 * 32 = 65536 tiles
  const int bb = tile >> 11;
  const int r2 = tile & 2047;
  const int yt = r2 >> 5, xt = r2 & 31;
  const int m0 = yt * 16, n0 = xt * 16;
  const int row = lane & 15, kh = lane >> 4;
  // A = mel[b] (Ty x 80); B[k][x] = mu_x[b,x,k] (i.e. mu_x rows are B columns)
  const float* Ap = mel + ((size_t)bb * TY + m0 + row) * NMEL + kh * 2;
  const float* Bp = mux + ((size_t)bb * TX + n0 + row) * NMEL + kh * 2;
  v8f acc = {};
  for (int k = 0; k < NMEL; k += 4) {
    v2f a, b;
    a.x = Ap[k]; a.y = Ap[k + 1];
    b.x = Bp[k]; b.y = Bp[k + 1];
    acc = wmma4(a, b, acc);
  }
  const int xl = xlens[bb], yl = ylens[bb];
  const int x = n0 + row;
  const float msq = musq[bb * TX + x];
  const float kconst = -0.5f * LOG2PI * (float)NMEL;
  for (int r = 0; r < 8; ++r) {
    int y = m0 + r + kh * 8;
    float v = acc[r] + ysq[bb * TY + y] + msq + kconst;
    bool valid = (x < xl) && (y < yl);
    LP[((size_t)bb * TY + y) * TX + x] = valid ? v : NEGF;
  }
}

// ---------------------------------------------------------------------------
// Kernel D: Viterbi DP + backtrack + mu_y gather + loss partials.
// One block per batch, 512 threads (thread == x). Direction bits (64 KB) kept
// in CDNA5's large LDS; one barrier per DP step via double-buffered exchange.
// ---------------------------------------------------------------------------
__global__ void k_dp(const float* __restrict__ LP, const float* __restrict__ logw,
                     const float* __restrict__ mux, const float* __restrict__ mel,
                     const int* __restrict__ xlens, const int* __restrict__ ylens,
                     float* __restrict__ outMu, float* __restrict__ parts) {
  extern __shared__ unsigned char smem[];
  unsigned* dirs = (unsigned*)smem;                               // TY*16 u32 = 64 KB
  float*    sbuf = (float*)(smem + TY * 16 * 4);                  // 2*TX floats
  int*      pidx = (int*)(smem + TY * 16 * 4 + 2 * TX * 4);       // TY ints
  int*      durc = (int*)((unsigned char*)pidx + TY * 4);         // TX ints
  float*    red  = (float*)((unsigned char*)durc + TX * 4);       // TX floats

  const int b   = blockIdx.x;
  const int tid = threadIdx.x;
  const int xl  = xlens[b], yl = ylens[b];
  const float* lpB = LP + (size_t)b * TY * TX;

  durc[tid] = 0;
  if (tid < 16) dirs[tid] = 0u;                                   // y = 0 row
  __syncthreads();

  // forward DP, v[x] in a register, prefetch next row
  float v  = lpB[tid] + (tid == 0 ? 0.f : NEGF);
  float lp = lpB[TX + tid];
  int par = 0;
  for (int y = 1; y < TY; ++y) {
    float lpn = (y < TY - 1) ? lpB[(size_t)(y + 1) * TX + tid] : 0.f;
    sbuf[par * TX + tid] = v;
    __syncthreads();
    float prev = (tid == 0) ? NEGF : sbuf[par * TX + tid - 1];
    bool diag = (prev >= v);
    v = lp + fmaxf(v, prev);
    unsigned msk = __builtin_amdgcn_ballot_w32(diag);
    if ((tid & 31) == 0) dirs[y * 16 + (tid >> 5)] = msk;
    lp = lpn;
    par ^= 1;
  }
  __syncthreads();

  // backtrack (inherently serial) from LDS
  if (tid == 0) {
    int idx = xl - 1;
    for (int y = TY - 1; y >= 0; --y) {
      bool act = (y < yl);
      pidx[y] = act ? idx : -1;
      if (act) {
        durc[idx] += 1;
        if (y > 0) {
          unsigned mm = dirs[y * 16 + (idx >> 5)];
          idx -= (int)((mm >> (idx & 31)) & 1u);
        }
      }
    }
  }
  __syncthreads();

  // duration-loss partial (deterministic: per-batch slot, no float atomics)
  float local = 0.f;
  if (tid < xl) {
    float lw_ = logf(1e-8f + (float)durc[tid]);
    float d = logw[b * TX + tid] - lw_;
    local = d * d;
  }
  red[tid] = local;
  __syncthreads();
  for (int s = TX / 2; s > 0; s >>= 1) {
    if (tid < s) red[tid] += red[tid + s];
    __syncthreads();
  }
  if (tid == 0) { parts[b] = red[0]; parts[32 + b] = (float)xl; }
  __syncthreads();

  // mu_y gather (hard monotonic path => gather, not GEMM) + prior partial
  float pl = 0.f;
  for (int e = tid; e < TY * NMEL; e += TX) {
    int y = e / NMEL, c = e - y * NMEL;
    float o = 0.f;
    if (y < yl) {
      int ix = pidx[y];
      float m = mux[((size_t)b * TX + ix) * NMEL + c];
      float d = mel[((size_t)b * TY + y) * NMEL + c] - m;
      pl += 0.5f * (d * d + LOG2PI);
      o = m;
    }
    outMu[((size_t)b * TY + y) * NMEL + c] = o;
  }
  __syncthreads();
  red[tid] = pl;
  __syncthreads();
  for (int s = TX / 2; s > 0; s >>= 1) {
    if (tid < s) red[tid] += red[tid + s];
    __syncthreads();
  }
  if (tid == 0) { parts[64 + b] = red[0]; parts[96 + b] = (float)yl; }
}

__global__ void k_final(const float* __restrict__ parts, float* __restrict__ outLoss) {
  float dn = 0.f, dx = 0.f, pn = 0.f, py = 0.f;
  for (int i = 0; i < B_; ++i) {
    dn += parts[i]; dx += parts[32 + i]; pn += parts[64 + i]; py += parts[96 + i];
  }
  *outLoss = dn / dx + pn / (py * (float)NMEL);
}

// ---------------------------------------------------------------------------
extern "C" void kernel_launch(void* const* d_in, const int* in_sizes, int n_in,
                              void* d_out, int out_size, void* d_ws, size_t ws_size,
                              hipStream_t stream) {
  const int*   speakers = (const int*)  d_in[0];
  const float* contents = (const float*)d_in[1];
  const int*   xlens    = (const int*)  d_in[2];
  const float* mel      = (const float*)d_in[3];
  const int*   ylens    = (const int*)  d_in[4];
  const float* W_text   = (const float*)d_in[5];
  const float* W_mel    = (const float*)d_in[6];
  const float* W_dur    = (const float*)d_in[7];
  const float* spk_emb  = (const float*)d_in[8];

  float* out = (float*)d_out;                 // [B*TY*NMEL] mu_y, then 1 scalar loss
  float* ws  = (float*)d_ws;

  size_t off = 0;
  float* feat = ws + off; off += (size_t)B_ * TX * DD;     // 16 MB
  float* mux  = ws + off; off += (size_t)B_ * TX * NMEL;   // 5 MB
  float* musq = ws + off; off += (size_t)B_ * TX;
  float* ysq  = ws + off; off += (size_t)B_ * TY;
  float* logw = ws + off; off += (size_t)B_ * TX;
  float* LP   = ws + off; off += (size_t)B_ * TY * TX;     // 64 MB (B,Ty,Tx)
  float* part = ws + off; off += 128;

  // GEMM 1: features
  k_features<<<2048, 256, 0, stream>>>(contents, W_text, speakers, spk_emb, xlens, feat);
  // GEMM 2: mu_x
  k_mux<<<640, 256, 0, stream>>>(feat, W_mel, xlens, mux);
  // row reductions
  k_rowops<<<(B_ * TX) / 256, 256, 0, stream>>>(feat, mux, W_dur, xlens, logw, musq);
  k_ysq<<<(B_ * TY) / 256, 256, 0, stream>>>(mel, ysq);
  // GEMM 3: log-prior (transposed layout for contiguous DP rows)
  k_logprior<<<8192, 256, 0, stream>>>(mel, mux, ysq, musq, xlens, ylens, LP);
  // Viterbi DP + backtrack + gather + partials (77,824 B dynamic LDS per WG)
  size_t smem = (size_t)TY * 16 * 4 + 2 * TX * 4 + TY * 4 + TX * 4 + TX * 4;
  k_dp<<<B_, TX, smem, stream>>>(LP, logw, mux, mel, xlens, ylens, out, part);
  // final scalar
  k_final<<<1, 1, 0, stream>>>(part, out + (size_t)B_ * TY * NMEL);
}